// Model_3470333575377
// MI455X (gfx1250) — compile-verified
//
#include <hip/hip_runtime.h>

typedef __attribute__((ext_vector_type(2))) float v2f;
typedef __attribute__((ext_vector_type(4))) float v4f;
typedef __attribute__((ext_vector_type(8))) float v8f;

namespace {
constexpr int kHeads         = 16;
constexpr int kTotalSeq      = 65536;
constexpr int kHeadDim       = 64;
constexpr int kBatch         = 32;
constexpr int kMaxSeq        = 4096;
constexpr int kTokensPerTile = 16;
constexpr int kTilesPerWave  = 4;
constexpr int kWavesPerBlock = 8;
constexpr int kThreadsPerBlock = kWavesPerBlock * 32;            // 256 = 8 wave32
constexpr int kTiles         = kTotalSeq / kTokensPerTile;       // 4096
constexpr int kJobsPerHead   = kTiles / kTilesPerWave;           // 1024 (pow2)
constexpr int kTotalJobs     = kHeads * kJobsPerHead;            // 16384 waves
constexpr int kBlocks        = kTotalJobs / kWavesPerBlock;      // 2048
}  // namespace

// ---------------------------------------------------------------------------
// Prologue: one validity byte per token (t - begin < MAX_SEQ_LEN), so the
// 512 MiB streaming kernel never touches the offset table. 64 KB result is
// L2-resident for the whole main kernel. The offset table is pulled into LDS
// with the gfx1250 async-to-LDS path.
// ---------------------------------------------------------------------------
__global__ __launch_bounds__(256)
void mask_kernel(const int* __restrict__ offsets, unsigned char* __restrict__ mask) {
  __shared__ int s_off[kBatch + 1];
  const int tid = threadIdx.x;

#if __has_builtin(__builtin_amdgcn_global_load_async_to_lds_b32)
  if (tid <= kBatch) {
    __builtin_amdgcn_global_load_async_to_lds_b32(
        (__attribute__((address_space(1))) int*)(offsets + tid),
        (__attribute__((address_space(3))) int*)&s_off[tid],
        /*offset=*/0, /*cpol=*/0);
  }
#if __has_builtin(__builtin_amdgcn_s_wait_asynccnt)
  __builtin_amdgcn_s_wait_asynccnt(0);
#else
  asm volatile("s_wait_asynccnt 0x0" ::: "memory");
#endif
#else
  if (tid <= kBatch) s_off[tid] = offsets[tid];
#endif
  __syncthreads();

  const int t = blockIdx.x * 256 + tid;
  // begin = largest offset <= t (offsets monotone, off[0]=0, off[32]=kTotalSeq)
  int lo = 0, hi = kBatch;
#pragma unroll
  for (int s = 0; s < 5; ++s) {
    const int  mid = (lo + hi) >> 1;
    const bool le  = (s_off[mid] <= t);
    lo = le ? mid : lo;
    hi = le ? hi : mid;
  }
  mask[t] = ((t - s_off[lo]) < kMaxSeq) ? (unsigned char)1 : (unsigned char)0;
}

// ---------------------------------------------------------------------------
// Main: rowsum over HEAD_DIM via per-lane f32 FMAs + one V_WMMA_F32_16X16X4_F32
// per 16-token tile (B == all-ones turns the WMMA into a lane-pair reduction
// plus row broadcast). Bandwidth-bound: 512 MiB NT-streamed once.
// ---------------------------------------------------------------------------
template <bool kUseMask>
__global__ __launch_bounds__(kThreadsPerBlock)
void rowsum_jagged_kernel(const float* __restrict__ o,
                          const float* __restrict__ dO,
                          const int* __restrict__ offsets,
                          const unsigned char* __restrict__ maskbuf,
                          float* __restrict__ out) {
  const int tid  = threadIdx.x;
  const int lane = tid & 31;
  const int wave = tid >> 5;

  // Fallback path keeps the offset table in LDS (unused, size-1 otherwise).
  __shared__ int s_off[kUseMask ? 1 : (kBatch + 1)];
  if constexpr (!kUseMask) {
    if (tid <= kBatch) s_off[tid] = offsets[tid];
    __syncthreads();
  }

  const int job = blockIdx.x * kWavesPerBlock + wave;
  const int h   = job >> 10;                 // job / kJobsPerHead (1024)
  const int tg  = job & (kJobsPerHead - 1);

  const int m    = lane & 15;   // token row within the 16-token tile
  const int half = lane >> 4;   // 0: dims 0..31, 1: dims 32..63

  const size_t head_base = (size_t)h * kTotalSeq * kHeadDim;
  const int    tile0     = tg * kTilesPerWave;
  const size_t lane_off  = (size_t)m * kHeadDim + (size_t)half * 32;

  // Store mapping (loop-invariant): lanes 0..7 -> rows 0..7, lanes 16..23 ->
  // rows 8..15; idle lanes get an in-tile row too (bounds-safe mask loads).
  const int  trow       = (lane & 7) | ((lane & 16) >> 1);
  const bool store_lane = (lane < 8) || (lane >= 16 && lane < 24);
  const int  idx        = lane & 7;  // which acc VGPR holds this lane's token

  const v2f ones = {1.0f, 1.0f};

  for (int it = 0; it < kTilesPerWave; ++it) {
    const int t0 = (tile0 + it) * kTokensPerTile;
    const int t  = t0 + trow;

    // Issue the (L2-hot) mask byte early so it overlaps the HBM stream.
    unsigned int mv = 1u;
    if constexpr (kUseMask) mv = maskbuf[t];

    const float* po = o  + head_base + (size_t)t0 * kHeadDim + lane_off;
    const float* pd = dO + head_base + (size_t)t0 * kHeadDim + lane_off;

    if (it + 1 < kTilesPerWave) {  // uniform branch: keep the stream ahead
      __builtin_prefetch(po + kTokensPerTile * kHeadDim, 0, 0);
      __builtin_prefetch(pd + kTokensPerTile * kHeadDim, 0, 0);
    }

    // Per-lane partial dots over 32 dims: 16 x global_load_b128 (NT).
    float s0 = 0.0f, s1 = 0.0f;
#pragma unroll
    for (int c = 0; c < 4; ++c) {
      v4f a0 = __builtin_nontemporal_load((const v4f*)(po + 4 * c));
      v4f b0 = __builtin_nontemporal_load((const v4f*)(pd + 4 * c));
      v4f a1 = __builtin_nontemporal_load((const v4f*)(po + 16 + 4 * c));
      v4f b1 = __builtin_nontemporal_load((const v4f*)(pd + 16 + 4 * c));
      s0 = fmaf(a0.x, b0.x, s0); s0 = fmaf(a0.y, b0.y, s0);
      s0 = fmaf(a0.z, b0.z, s0); s0 = fmaf(a0.w, b0.w, s0);
      s1 = fmaf(a1.x, b1.x, s1); s1 = fmaf(a1.y, b1.y, s1);
      s1 = fmaf(a1.z, b1.z, s1); s1 = fmaf(a1.w, b1.w, s1);
    }

    // One WMMA per tile: D[i][j] = s0(i)+s1(i)+s0(i+16)+s1(i+16) = rowsum(t0+i).
    v2f amat;
    amat[0] = s0;
    amat[1] = s1;
    v8f acc = {};
    acc = __builtin_amdgcn_wmma_f32_16x16x4_f32(
        /*neg_a=*/false, amat, /*neg_b=*/false, ones,
        /*c_mod=*/(short)0, acc, /*reuse_a=*/false, /*reuse_b=*/false);

    // acc[r]: lanes 0..15 hold rowsum(t0+r); lanes 16..31 hold rowsum(t0+r+8).
    float v = acc[0];
#pragma unroll
    for (int r = 1; r < 8; ++r) v = (idx == r) ? (float)acc[r] : v;

    if constexpr (!kUseMask) {
      int lo = 0, hi = kBatch;
#pragma unroll
      for (int s = 0; s < 5; ++s) {
        const int  mid = (lo + hi) >> 1;
        const bool le  = (s_off[mid] <= t);
        lo = le ? mid : lo;
        hi = le ? hi : mid;
      }
      mv = ((t - s_off[lo]) < kMaxSeq) ? 1u : 0u;
    }

    if (store_lane) {
      __builtin_nontemporal_store(mv ? v : 0.0f,
                                  out + (size_t)h * kTotalSeq + t);
    }
  }
}

extern "C" void kernel_launch(void* const* d_in, const int* in_sizes, int n_in,
                              void* d_out, int out_size, void* d_ws, size_t ws_size,
                              hipStream_t stream) {
  const float* o   = (const float*)d_in[0];
  const float* dO  = (const float*)d_in[1];
  const int*   off = (const int*)d_in[2];
  float*       out = (float*)d_out;
  (void)in_sizes; (void)n_in; (void)out_size;

  if (ws_size >= (size_t)kTotalSeq) {
    unsigned char* mask = (unsigned char*)d_ws;
    mask_kernel<<<kTotalSeq / 256, 256, 0, stream>>>(off, mask);
    rowsum_jagged_kernel<true><<<kBlocks, kThreadsPerBlock, 0, stream>>>(
        o, dO, off, mask, out);
  } else {
    rowsum_jagged_kernel<false><<<kBlocks, kThreadsPerBlock, 0, stream>>>(
        o, dO, off, nullptr, out);
  }
}